// PennylaneLayer_20478404067593
// MI455X (gfx1250) — compile-verified
//
#include <hip/hip_runtime.h>
#include <math.h>

// 12 qubits -> 64x64 complex state matrix per sample.
// rows  = qubits 0..5  (qubit q -> bit (5-q) of row index)
// cols  = qubits 6..11 (qubit q -> bit (11-q) of col index)
#define LDA 65          // padded LDS row stride (floats) to break bank conflicts
#define NL  4

typedef __attribute__((ext_vector_type(2))) float v2f;
typedef __attribute__((ext_vector_type(8))) float v8f;

__device__ __forceinline__ float2 cmul(float2 a, float2 b) {
  return make_float2(a.x * b.x - a.y * b.y, a.x * b.y + a.y * b.x);
}
__device__ __forceinline__ float2 cadd(float2 a, float2 b) {
  return make_float2(a.x + b.x, a.y + b.y);
}

struct C2 { float2 a00, a01, a10, a11; };
__device__ __forceinline__ C2 mm2(const C2 L, const C2 R) {
  C2 o;
  o.a00 = cadd(cmul(L.a00, R.a00), cmul(L.a01, R.a10));
  o.a01 = cadd(cmul(L.a00, R.a01), cmul(L.a01, R.a11));
  o.a10 = cadd(cmul(L.a10, R.a00), cmul(L.a11, R.a10));
  o.a11 = cadd(cmul(L.a10, R.a01), cmul(L.a11, R.a11));
  return o;
}

// PennyLane Rot(phi,theta,omega) = RZ(omega) RY(theta) RZ(phi)
__device__ __forceinline__ void rot_gate(float phi, float th, float om,
                                         float2* u00, float2* u01,
                                         float2* u10, float2* u11) {
  float ct = cosf(0.5f * th), st = sinf(0.5f * th);
  float ap = -0.5f * (phi + om);  // angle of m00
  float am = 0.5f * (phi - om);   // angle of m01
  *u00 = make_float2(ct * cosf(ap), ct * sinf(ap));
  *u01 = make_float2(-st * cosf(am), -st * sinf(am));
  *u10 = make_float2(st * cosf(-am), st * sinf(-am));
  *u11 = make_float2(ct * cosf(-ap), ct * sinf(-ap));
}

// Apply controlled-U (left-mult) to 64x64 complex matrix in LDS.
__device__ __forceinline__ void apply_cgate(float* Mr, float* Mi,
                                            float2 u00, float2 u01,
                                            float2 u10, float2 u11,
                                            int cb, int tb, int tid) {
  for (int idx = tid; idx < 4096; idx += 256) {
    int i = idx >> 6, j = idx & 63;
    if ((i & cb) && !(i & tb)) {
      int a0 = i * LDA + j, a1 = (i | tb) * LDA + j;
      float2 x0 = make_float2(Mr[a0], Mi[a0]);
      float2 x1 = make_float2(Mr[a1], Mi[a1]);
      float2 y0 = cadd(cmul(u00, x0), cmul(u01, x1));
      float2 y1 = cadd(cmul(u10, x0), cmul(u11, x1));
      Mr[a0] = y0.x; Mi[a0] = y0.y;
      Mr[a1] = y1.x; Mi[a1] = y1.y;
    }
  }
}

// ---------------------------------------------------------------------------
// Kernel 1: precompute per-layer shared 64x64 matrices into d_ws.
// Per layer l (float2, row-major 64x64 each):
//   slot 0: MA  = CR(4,5)..CR(0,1) * U3High        (left-mult)
//   slot 1: BL  = U3Low^T                          (right-mult)
//   slot 2: MBT = (CR(4,5)..CR(0,1) low chain)^T   (right-mult)
// Boundary CRots (5,6) and (11,0) are U (x) I32 -> applied elementwise in
// the main kernel instead of as dense GEMMs.
// ---------------------------------------------------------------------------
__global__ __launch_bounds__(256) void qk_prep(const float* __restrict__ rotp,
                                               const float* __restrict__ entp,
                                               float2* __restrict__ W) {
  __shared__ float Mr[64 * LDA], Mi[64 * LDA];
  __shared__ float2 U[24][2][2];  // 0..11 = Rot params, 12..23 = enta params
  const int l = blockIdx.x;
  const int tid = threadIdx.x;

  if (tid < 24) {
    const float* p = (tid < 12) ? &rotp[(l * 12 + tid) * 3]
                                : &entp[(l * 12 + (tid - 12)) * 3];
    rot_gate(p[0], p[1], p[2], &U[tid][0][0], &U[tid][0][1],
             &U[tid][1][0], &U[tid][1][1]);
  }
  __syncthreads();

  float2* Wl = W + (size_t)l * 3 * 4096;

  // ---- MA: build U3High kron, then apply high CRot chain ----
  for (int idx = tid; idx < 4096; idx += 256) {
    int i = idx >> 6, j = idx & 63;
    float2 p = U[0][(i >> 5) & 1][(j >> 5) & 1];
    for (int q = 1; q < 6; q++)
      p = cmul(p, U[q][(i >> (5 - q)) & 1][(j >> (5 - q)) & 1]);
    Mr[i * LDA + j] = p.x; Mi[i * LDA + j] = p.y;
  }
  __syncthreads();
  for (int c = 0; c < 5; c++) {  // CRot(c, c+1), qubits 0..5
    apply_cgate(Mr, Mi, U[12 + c][0][0], U[12 + c][0][1],
                U[12 + c][1][0], U[12 + c][1][1],
                1 << (5 - c), 1 << (4 - c), tid);
    __syncthreads();
  }
  for (int idx = tid; idx < 4096; idx += 256) {
    int i = idx >> 6, j = idx & 63;
    Wl[0 * 4096 + idx] = make_float2(Mr[i * LDA + j], Mi[i * LDA + j]);
  }

  // ---- BL = U3Low^T (direct formula) ----
  for (int idx = tid; idx < 4096; idx += 256) {
    int i = idx >> 6, j = idx & 63;
    float2 p = U[6][(j >> 5) & 1][(i >> 5) & 1];  // transpose: [j][i]
    for (int q = 1; q < 6; q++)
      p = cmul(p, U[6 + q][(j >> (5 - q)) & 1][(i >> (5 - q)) & 1]);
    Wl[1 * 4096 + idx] = p;
  }
  __syncthreads();  // done reading Mr/Mi (MA)

  // ---- MBT: identity, apply low CRot chain, store transpose ----
  for (int idx = tid; idx < 4096; idx += 256) {
    int i = idx >> 6, j = idx & 63;
    Mr[i * LDA + j] = (i == j) ? 1.f : 0.f;
    Mi[i * LDA + j] = 0.f;
  }
  __syncthreads();
  for (int c = 0; c < 5; c++) {  // CRot(6+c, 7+c) -> sub-wires (c, c+1)
    apply_cgate(Mr, Mi, U[18 + c][0][0], U[18 + c][0][1],
                U[18 + c][1][0], U[18 + c][1][1],
                1 << (5 - c), 1 << (4 - c), tid);
    __syncthreads();
  }
  for (int idx = tid; idx < 4096; idx += 256) {
    int i = idx >> 6, j = idx & 63;
    Wl[2 * 4096 + idx] = make_float2(Mr[j * LDA + i], Mi[j * LDA + i]);
  }
}

// ---------------------------------------------------------------------------
// Karatsuba complex GEMM: D = A * B (64x64x64 complex f32, LDS-resident).
// 8 waves; wave w owns two 16x16 tiles (m_tile = w&3, n tiles 2*(w>>2)+{0,1}).
// 3 real WMMAs per tile per k-step:
//   P1 += Ar*Br, P2 += Ai*Bi, P3 += (Ar+Ai)*(Br+Bi)
//   Cr = P1 - P2, Ci = P3 - P1 - P2
// ---------------------------------------------------------------------------
__device__ __forceinline__ v8f wmma4(v2f a, v2f b, v8f c) {
  return __builtin_amdgcn_wmma_f32_16x16x4_f32(
      false, a, false, b, (short)0, c, false, false);
}

__device__ __forceinline__ void cgemm64(const float* Ar, const float* Ai,
                                        const float* Br, const float* Bi,
                                        float* Dr, float* Di) {
  const int tid = threadIdx.x;
  const int lane = tid & 31;
  const int w = tid >> 5;
  const int mt = (w & 3) * 16;
  const int nt0 = (w >> 2) * 32;
  const int nt1 = nt0 + 16;
  const int ml = lane & 15;        // A row-in-tile / B,C col-in-tile
  const int kb = (lane >> 4) * 2;  // K sub-offset for A/B fragments
  const int hi = lane >> 4;

  v8f p1_0 = {}, p2_0 = {}, p3_0 = {};
  v8f p1_1 = {}, p2_1 = {}, p3_1 = {};
  for (int k = 0; k < 64; k += 4) {
    const float* ap = Ar + (mt + ml) * LDA + k + kb;
    const float* aip = Ai + (mt + ml) * LDA + k + kb;
    v2f ar; ar.x = ap[0];  ar.y = ap[1];
    v2f ai; ai.x = aip[0]; ai.y = aip[1];
    v2f as = ar + ai;
    v2f br0, bi0, br1, bi1;
    br0.x = Br[(k + kb) * LDA + nt0 + ml]; br0.y = Br[(k + kb + 1) * LDA + nt0 + ml];
    bi0.x = Bi[(k + kb) * LDA + nt0 + ml]; bi0.y = Bi[(k + kb + 1) * LDA + nt0 + ml];
    br1.x = Br[(k + kb) * LDA + nt1 + ml]; br1.y = Br[(k + kb + 1) * LDA + nt1 + ml];
    bi1.x = Bi[(k + kb) * LDA + nt1 + ml]; bi1.y = Bi[(k + kb + 1) * LDA + nt1 + ml];
    v2f bs0 = br0 + bi0;
    v2f bs1 = br1 + bi1;

    p1_0 = wmma4(ar, br0, p1_0);
    p2_0 = wmma4(ai, bi0, p2_0);
    p3_0 = wmma4(as, bs0, p3_0);
    p1_1 = wmma4(ar, br1, p1_1);
    p2_1 = wmma4(ai, bi1, p2_1);
    p3_1 = wmma4(as, bs1, p3_1);
  }

  v8f cr0 = p1_0 - p2_0;
  v8f ci0 = p3_0 - p1_0 - p2_0;
  v8f cr1 = p1_1 - p2_1;
  v8f ci1 = p3_1 - p1_1 - p2_1;

  for (int v = 0; v < 8; v++) {
    int m = mt + v + 8 * hi;
    int n0 = nt0 + ml, n1 = nt1 + ml;
    Dr[m * LDA + n0] = cr0[v]; Di[m * LDA + n0] = ci0[v];
    Dr[m * LDA + n1] = cr1[v]; Di[m * LDA + n1] = ci1[v];
  }
}

// ---------------------------------------------------------------------------
// Kernel 2: one workgroup (256 threads / 8 waves) per sample.
// ---------------------------------------------------------------------------
__global__ __launch_bounds__(256) void qk_main(const float* __restrict__ x,
                                               const float* __restrict__ entp,
                                               const float2* __restrict__ W,
                                               float* __restrict__ out) {
  __shared__ float S0r[64 * LDA], S0i[64 * LDA];
  __shared__ float S1r[64 * LDA], S1i[64 * LDA];
  __shared__ float OpR[64 * LDA], OpI[64 * LDA];
  __shared__ float2 E[12][2][2];     // per-qubit encoding 2x2s (layer-indep)
  __shared__ float2 K8[4][64];       // 3-qubit sub-Krons of E
  __shared__ float2 CU[NL][2][2][2]; // boundary CRot gates: [l][0]=wire5, [1]=wire11

  const int tid = threadIdx.x;
  const int s = blockIdx.x;

  // Per-qubit encoding gate: RY(x3) RZ(x2) RX(x1) RY(x0)
  if (tid < 12) {
    const float* xs = x + (size_t)s * 48;
    float a = xs[0 * 12 + tid], b = xs[1 * 12 + tid];
    float c = xs[2 * 12 + tid], d = xs[3 * 12 + tid];
    float ca = cosf(0.5f * a), sa = sinf(0.5f * a);
    float cb = cosf(0.5f * b), sb = sinf(0.5f * b);
    float cc = cosf(0.5f * c), sc = sinf(0.5f * c);
    float cd = cosf(0.5f * d), sd = sinf(0.5f * d);
    C2 M  = { {ca, 0.f}, {-sa, 0.f}, {sa, 0.f}, {ca, 0.f} };  // RY(a)
    C2 RX = { {cb, 0.f}, {0.f, -sb}, {0.f, -sb}, {cb, 0.f} };
    C2 RZ = { {cc, -sc}, {0.f, 0.f}, {0.f, 0.f}, {cc, sc} };
    C2 RY = { {cd, 0.f}, {-sd, 0.f}, {sd, 0.f}, {cd, 0.f} };
    M = mm2(RX, M); M = mm2(RZ, M); M = mm2(RY, M);
    E[tid][0][0] = M.a00; E[tid][0][1] = M.a01;
    E[tid][1][0] = M.a10; E[tid][1][1] = M.a11;
  }
  // Boundary CRot gate matrices per layer (wires 5 and 11)
  if (tid < 2 * NL) {
    int l = tid >> 1, which = tid & 1;
    const float* p = &entp[(l * 12 + (which ? 11 : 5)) * 3];
    rot_gate(p[0], p[1], p[2], &CU[l][which][0][0], &CU[l][which][0][1],
             &CU[l][which][1][0], &CU[l][which][1][1]);
  }
  // State |0...0>: S[0][0] = 1
  for (int idx = tid; idx < 4096; idx += 256) {
    int i = idx >> 6, j = idx & 63;
    S0r[i * LDA + j] = (idx == 0) ? 1.f : 0.f;
    S0i[i * LDA + j] = 0.f;
  }
  __syncthreads();

  // 3-qubit sub-Krons: K8[g] = E[3g] (x) E[3g+1] (x) E[3g+2], one entry/thread
  {
    int g = tid >> 6, a = (tid >> 3) & 7, b = tid & 7;
    float2 p = cmul(cmul(E[3 * g][(a >> 2) & 1][(b >> 2) & 1],
                         E[3 * g + 1][(a >> 1) & 1][(b >> 1) & 1]),
                    E[3 * g + 2][a & 1][b & 1]);
    K8[g][a * 8 + b] = p;
  }
  __syncthreads();

  float *curR = S0r, *curI = S0i, *nxtR = S1r, *nxtI = S1i;

  for (int l = 0; l < NL; l++) {
    const float2* Wl = W + (size_t)l * 3 * 4096;

    // (1) S <- EncHigh * S     (EncHigh[i][j] = K8[0] * K8[1], one cmul/entry)
    for (int idx = tid; idx < 4096; idx += 256) {
      int i = idx >> 6, j = idx & 63;
      float2 p = cmul(K8[0][((i >> 3) << 3) | (j >> 3)],
                      K8[1][((i & 7) << 3) | (j & 7)]);
      OpR[i * LDA + j] = p.x; OpI[i * LDA + j] = p.y;
    }
    __syncthreads();
    cgemm64(OpR, OpI, curR, curI, nxtR, nxtI);
    __syncthreads();
    { float* t; t = curR; curR = nxtR; nxtR = t; t = curI; curI = nxtI; nxtI = t; }

    // (2) S <- S * EncLow^T
    for (int idx = tid; idx < 4096; idx += 256) {
      int i = idx >> 6, j = idx & 63;
      float2 p = cmul(K8[2][((j >> 3) << 3) | (i >> 3)],
                      K8[3][((j & 7) << 3) | (i & 7)]);
      OpR[i * LDA + j] = p.x; OpI[i * LDA + j] = p.y;
    }
    __syncthreads();
    cgemm64(curR, curI, OpR, OpI, nxtR, nxtI);
    __syncthreads();
    { float* t; t = curR; curR = nxtR; nxtR = t; t = curI; curI = nxtI; nxtI = t; }

    // (3) S <- MA * S
    for (int idx = tid; idx < 4096; idx += 256) {
      float2 v = Wl[idx];
      int i = idx >> 6, j = idx & 63;
      OpR[i * LDA + j] = v.x; OpI[i * LDA + j] = v.y;
    }
    __syncthreads();
    cgemm64(OpR, OpI, curR, curI, nxtR, nxtI);
    __syncthreads();
    { float* t; t = curR; curR = nxtR; nxtR = t; t = curI; curI = nxtI; nxtI = t; }

    // (4) S <- S * BL
    for (int idx = tid; idx < 4096; idx += 256) {
      float2 v = Wl[4096 + idx];
      int i = idx >> 6, j = idx & 63;
      OpR[i * LDA + j] = v.x; OpI[i * LDA + j] = v.y;
    }
    __syncthreads();
    cgemm64(curR, curI, OpR, OpI, nxtR, nxtI);
    __syncthreads();
    { float* t; t = curR; curR = nxtR; nxtR = t; t = curI; curI = nxtI; nxtI = t; }

    // (5) stage MBT; and in parallel apply CRot(5,6) elementwise in place:
    //     odd rows (control qubit 5 = row LSB), target = col MSB pairs.
    for (int idx = tid; idx < 4096; idx += 256) {
      float2 v = Wl[2 * 4096 + idx];
      int i = idx >> 6, j = idx & 63;
      OpR[i * LDA + j] = v.x; OpI[i * LDA + j] = v.y;
    }
    {
      float2 u00 = CU[l][0][0][0], u01 = CU[l][0][0][1];
      float2 u10 = CU[l][0][1][0], u11 = CU[l][0][1][1];
      for (int t = tid; t < 1024; t += 256) {
        int i = ((t >> 5) << 1) | 1;  // odd row
        int jl = t & 31;
        int a0 = i * LDA + jl, a1 = a0 + 32;
        float2 x0 = make_float2(curR[a0], curI[a0]);
        float2 x1 = make_float2(curR[a1], curI[a1]);
        float2 y0 = cadd(cmul(u00, x0), cmul(u01, x1));
        float2 y1 = cadd(cmul(u10, x0), cmul(u11, x1));
        curR[a0] = y0.x; curI[a0] = y0.y;
        curR[a1] = y1.x; curI[a1] = y1.y;
      }
    }
    __syncthreads();

    // (6) S <- S * MBT
    cgemm64(curR, curI, OpR, OpI, nxtR, nxtI);
    __syncthreads();
    { float* t; t = curR; curR = nxtR; nxtR = t; t = curI; curI = nxtI; nxtI = t; }

    // (7) CRot(11,0) elementwise in place: odd cols (control qubit 11 =
    //     col LSB), target = row MSB pairs. Next iteration's staging + sync
    //     (or the post-loop sync) orders this before the next GEMM.
    {
      float2 u00 = CU[l][1][0][0], u01 = CU[l][1][0][1];
      float2 u10 = CU[l][1][1][0], u11 = CU[l][1][1][1];
      for (int t = tid; t < 1024; t += 256) {
        int il = t >> 5;              // 0..31
        int j = ((t & 31) << 1) | 1;  // odd col
        int a0 = il * LDA + j, a1 = (il + 32) * LDA + j;
        float2 x0 = make_float2(curR[a0], curI[a0]);
        float2 x1 = make_float2(curR[a1], curI[a1]);
        float2 y0 = cadd(cmul(u00, x0), cmul(u01, x1));
        float2 y1 = cadd(cmul(u10, x0), cmul(u11, x1));
        curR[a0] = y0.x; curI[a0] = y0.y;
        curR[a1] = y1.x; curI[a1] = y1.y;
      }
    }
  }
  __syncthreads();

  // <Z_q> = sum_n |amp_n|^2 * (1 - 2*bit_q(n)), bit_q(n) = (n >> (11-q)) & 1
  float z[12];
  for (int q = 0; q < 12; q++) z[q] = 0.f;
  for (int idx = tid; idx < 4096; idx += 256) {
    int i = idx >> 6, j = idx & 63;
    float pr = curR[i * LDA + j], pi = curI[i * LDA + j];
    float p = pr * pr + pi * pi;
    for (int q = 0; q < 12; q++)
      z[q] += ((idx >> (11 - q)) & 1) ? -p : p;
  }
  __syncthreads();
  float* red = OpR;  // reuse: 256*12 = 3072 <= 64*65
  for (int q = 0; q < 12; q++) red[tid * 12 + q] = z[q];
  __syncthreads();
  for (int sft = 128; sft > 0; sft >>= 1) {
    if (tid < sft)
      for (int q = 0; q < 12; q++)
        red[tid * 12 + q] += red[(tid + sft) * 12 + q];
    __syncthreads();
  }
  if (tid < 12) out[(size_t)s * 12 + tid] = red[tid];
}

extern "C" void kernel_launch(void* const* d_in, const int* in_sizes, int n_in,
                              void* d_out, int out_size, void* d_ws, size_t ws_size,
                              hipStream_t stream) {
  (void)n_in; (void)out_size; (void)ws_size;
  const float* x    = (const float*)d_in[0];  // (B, 4, 12) f32
  const float* rotp = (const float*)d_in[1];  // (4, 12, 3) f32
  const float* entp = (const float*)d_in[2];  // (4, 12, 3) f32
  float2* W = (float2*)d_ws;                  // 4 layers * 3 mats * 4096 float2 = 384 KB
  const int B = in_sizes[0] / 48;

  qk_prep<<<NL, 256, 0, stream>>>(rotp, entp, W);
  qk_main<<<B, 256, 0, stream>>>(x, entp, W, (float*)d_out);
}